// Attention_80487687127519
// MI455X (gfx1250) — compile-verified
//
#include <hip/hip_runtime.h>
#include <math.h>

typedef __attribute__((ext_vector_type(2))) float v2f;
typedef __attribute__((ext_vector_type(8))) float v8f;

#define B_    128
#define T_    2000
#define DQ_   1024
#define DMEM_ 512
#define DATT_ 128

// ---------------------------------------------------------------------------
// Kernel 1: pq[b][a] = sum_q H[b][q] * Wq[a][q]   (128x128 = 64 WMMA tiles)
// Uses V_WMMA_F32_16X16X4_F32. One wave per 16x16 output tile, K stepped by 4.
// f32 A-frag (16x4): lanes 0-15 hold K={0,1}, lanes 16-31 hold K={2,3}
// f32 B-frag (4x16): VGPR0 = rows K0/K2 across lanes, VGPR1 = rows K1/K3
// ---------------------------------------------------------------------------
__global__ __launch_bounds__(128) void wmma_proj_kernel(
    const float* __restrict__ H, const float* __restrict__ Wq,
    float* __restrict__ pq) {
  const int wib  = threadIdx.x >> 5;            // wave in block (0..3)
  const int lane = threadIdx.x & 31;
  const int wid  = blockIdx.x * 4 + wib;        // 0..63
  const int tm   = wid >> 3;                    // batch tile
  const int tn   = wid & 7;                     // att-dim tile
  const int half = lane >> 4;
  const int lrow = lane & 15;

  const float* __restrict__ Arow = H  + (size_t)(tm * 16 + lrow) * DQ_;
  const float* __restrict__ Brow = Wq + (size_t)(tn * 16 + lrow) * DQ_;

  v8f c = {};
  for (int k = 0; k < DQ_; k += 4) {
    const int kk = k + 2 * half;                // contiguous float2 along K
    v2f a, b;
    a.x = Arow[kk]; a.y = Arow[kk + 1];
    b.x = Brow[kk]; b.y = Brow[kk + 1];
    c = __builtin_amdgcn_wmma_f32_16x16x4_f32(
        /*neg_a=*/false, a, /*neg_b=*/false, b,
        /*c_mod=*/(short)0, c, /*reuse_a=*/false, /*reuse_b=*/false);
  }
  // C/D layout: VGPR r, lane half h -> M = r + 8*h, N = lane%16
#pragma unroll
  for (int r = 0; r < 8; ++r) {
    const int m = tm * 16 + r + 8 * half;
    const int n = tn * 16 + lrow;
    pq[m * DATT_ + n] = c[r];
  }
}

// ---------------------------------------------------------------------------
// Kernel 2: E[b][t] = mask ? -inf : v . tanh(pq[b] + pm[b][t])
// One wave per t; each lane owns 4 of the 128 att dims (float4 loads).
// ---------------------------------------------------------------------------
__global__ __launch_bounds__(256) void energy_kernel(
    const float* __restrict__ pm, const float* __restrict__ pq,
    const float* __restrict__ v, const unsigned char* __restrict__ mask,
    float* __restrict__ E) {
  const int b    = blockIdx.x;
  const int wid  = threadIdx.x >> 5;
  const int lane = threadIdx.x & 31;
  const int t    = blockIdx.y * 8 + wid;        // gridDim.y*8 == T_

  const float4 pmv = ((const float4*)(pm + ((size_t)b * T_ + t) * DATT_))[lane];
  const float4 pqv = ((const float4*)(pq + (size_t)b * DATT_))[lane];
  const float4 vv  = ((const float4*)v)[lane];

  float e = vv.x * tanhf(pqv.x + pmv.x) + vv.y * tanhf(pqv.y + pmv.y)
          + vv.z * tanhf(pqv.z + pmv.z) + vv.w * tanhf(pqv.w + pmv.w);
#pragma unroll
  for (int off = 16; off >= 1; off >>= 1) e += __shfl_xor(e, off, 32);

  if (lane == 0) {
    const size_t idx = (size_t)b * T_ + t;
    E[idx] = mask[idx] ? -__builtin_inff() : e;
  }
}

// ---------------------------------------------------------------------------
// Kernel 3: per-row softmax stats: rowmax[b], rowsum[b] = sum exp(E - max)
// ---------------------------------------------------------------------------
__global__ __launch_bounds__(256) void softmax_stats_kernel(
    const float* __restrict__ E, float* __restrict__ rowmax,
    float* __restrict__ rowsum) {
  const int b = blockIdx.x, tid = threadIdx.x;
  const float* __restrict__ Er = E + (size_t)b * T_;
  float vals[8];
  float m = -__builtin_inff();
#pragma unroll
  for (int i = 0; i < 8; ++i) {
    const int t = tid + i * 256;
    vals[i] = (t < T_) ? Er[t] : -__builtin_inff();
    m = fmaxf(m, vals[i]);
  }
  __shared__ float red[256];
  red[tid] = m; __syncthreads();
  for (int s = 128; s >= 1; s >>= 1) {
    if (tid < s) red[tid] = fmaxf(red[tid], red[tid + s]);
    __syncthreads();
  }
  m = red[0]; __syncthreads();

  float sum = 0.f;
#pragma unroll
  for (int i = 0; i < 8; ++i) {
    const int t = tid + i * 256;
    if (t < T_) sum += expf(vals[i] - m);      // exp(-inf - m) == 0
  }
  red[tid] = sum; __syncthreads();
  for (int s = 128; s >= 1; s >>= 1) {
    if (tid < s) red[tid] += red[tid + s];
    __syncthreads();
  }
  if (tid == 0) { rowmax[b] = m; rowsum[b] = red[0]; }
}

// ---------------------------------------------------------------------------
// Kernel 4: weights[b][t] = exp(E - rowmax) / rowsum  -> output region 2
// ---------------------------------------------------------------------------
__global__ __launch_bounds__(256) void weights_kernel(
    const float* __restrict__ E, const float* __restrict__ rowmax,
    const float* __restrict__ rowsum, float* __restrict__ w) {
  const int idx = blockIdx.x * 256 + threadIdx.x;
  if (idx >= B_ * T_) return;
  const int b = idx / T_;
  w[idx] = expf(E[idx] - rowmax[b]) / rowsum[b];
}

// ---------------------------------------------------------------------------
// Kernel 5: ctx[b][d] = sum_t w[b][t] * memory[b][t][d]
// Dominant traffic (524 MB). Weights staged in LDS, float2 streaming loads,
// prefetch ahead (global_prefetch_b8). grid = (2 col-chunks, 128 batches).
// ---------------------------------------------------------------------------
__global__ __launch_bounds__(128) void context_kernel(
    const float* __restrict__ mem, const float* __restrict__ w,
    float* __restrict__ ctx) {
  const int b   = blockIdx.y;
  const int d   = blockIdx.x * 256 + threadIdx.x * 2;   // 2 chunks x 256 cols
  const int tid = threadIdx.x;

  __shared__ float wsm[128];
  float2 acc = make_float2(0.f, 0.f);
  const float* __restrict__ wrow = w + (size_t)b * T_;
  const float* __restrict__ mcol = mem + (size_t)b * T_ * DMEM_ + d;

  for (int tb = 0; tb < T_; tb += 128) {
    const int t = tb + tid;
    wsm[tid] = (t < T_) ? wrow[t] : 0.f;
    __syncthreads();
    const int n = (T_ - tb < 128) ? (T_ - tb) : 128;
    for (int tt = 0; tt < n; ++tt) {
      const float2 mv = *(const float2*)(mcol + (size_t)(tb + tt) * DMEM_);
      __builtin_prefetch(mcol + (size_t)(tb + tt + 8) * DMEM_, 0, 0);
      acc.x = fmaf(wsm[tt], mv.x, acc.x);
      acc.y = fmaf(wsm[tt], mv.y, acc.y);
    }
    __syncthreads();
  }
  ctx[(size_t)b * DMEM_ + d]     = acc.x;
  ctx[(size_t)b * DMEM_ + d + 1] = acc.y;
}

// ---------------------------------------------------------------------------
extern "C" void kernel_launch(void* const* d_in, const int* in_sizes, int n_in,
                              void* d_out, int out_size, void* d_ws,
                              size_t ws_size, hipStream_t stream) {
  const float*         H    = (const float*)d_in[0];         // [128,1024]
  const float*         mem  = (const float*)d_in[1];         // [128,2000,512]
  const float*         pm   = (const float*)d_in[2];         // [128,2000,128]
  /* d_in[3] attention_weights_cat: unused by the reference */
  const unsigned char* mask = (const unsigned char*)d_in[4]; // [128,2000] bool
  const float*         Wq   = (const float*)d_in[5];         // [128,1024]
  const float*         v    = (const float*)d_in[6];         // [1,128]

  float* out = (float*)d_out;
  float* ctx = out;                      // [128,512]
  float* wts = out + B_ * DMEM_;         // [128,2000]

  float* ws     = (float*)d_ws;
  float* pq     = ws;                    // 16384 floats
  float* E      = ws + B_ * DATT_;       // 256000 floats
  float* rowmax = E + (size_t)B_ * T_;   // 128
  float* rowsum = rowmax + B_;           // 128

  wmma_proj_kernel<<<16, 128, 0, stream>>>(H, Wq, pq);
  energy_kernel<<<dim3(B_, T_ / 8), 256, 0, stream>>>(pm, pq, v, mask, E);
  softmax_stats_kernel<<<B_, 256, 0, stream>>>(E, rowmax, rowsum);
  weights_kernel<<<(B_ * T_) / 256, 256, 0, stream>>>(E, rowmax, rowsum, wts);
  context_kernel<<<dim3(2, B_), 128, 0, stream>>>(mem, wts, ctx);
}